// kmeans_distribution_layer_80745385165350
// MI455X (gfx1250) — compile-verified
//
#include <hip/hip_runtime.h>
#include <math.h>

#define B_N 16384
#define K_N 64
#define D_N 128

typedef __attribute__((ext_vector_type(2))) float v2f;
typedef __attribute__((ext_vector_type(8))) float v8f;
typedef __attribute__((__vector_size__(2 * sizeof(int)))) int v2i;

// ---- async Global->LDS helpers (gfx1250) --------------------------------
__device__ __forceinline__ void async_copy8(const float* gsrc, float* ldst) {
#if __has_builtin(__builtin_amdgcn_global_load_async_to_lds_b64)
    __builtin_amdgcn_global_load_async_to_lds_b64(
        (__attribute__((address_space(1))) v2i*)(size_t)gsrc,
        (__attribute__((address_space(3))) v2i*)(unsigned)(size_t)ldst,
        0, 0);
#else
    unsigned l = (unsigned)(size_t)ldst;
    unsigned long long g = (unsigned long long)(size_t)gsrc;
    asm volatile("global_load_async_to_lds_b64 %0, %1, off"
                 :: "v"(l), "v"(g) : "memory");
#endif
}

__device__ __forceinline__ void wait_async0() {
#if __has_builtin(__builtin_amdgcn_s_wait_asynccnt)
    __builtin_amdgcn_s_wait_asynccnt(0);
#else
    asm volatile("s_wait_asynccnt 0x0" ::: "memory");
#endif
}
__device__ __forceinline__ void wait_async1() {
#if __has_builtin(__builtin_amdgcn_s_wait_asynccnt)
    __builtin_amdgcn_s_wait_asynccnt(1);
#else
    asm volatile("s_wait_asynccnt 0x1" ::: "memory");
#endif
}

// ---------------------------------------------------------------- init ----
__global__ void __launch_bounds__(64)
gmm_init_kernel(float* weight, int* cursor, float* out_scalars) {
    int t = threadIdx.x;
    if (t < K_N) { weight[t] = 0.0f; cursor[t] = 0; }
    if (t == 0) { out_scalars[0] = 0.0f; out_scalars[1] = 0.0f; }
}

// ------------------------------------------------------------- norms -----
__global__ void __launch_bounds__(256)
gmm_norms_kernel(const float* __restrict__ x, const float* __restrict__ c,
                 float* __restrict__ xnorm, float* __restrict__ cnorm) {
    const int wave = threadIdx.x >> 5;
    const int lane = threadIdx.x & 31;
    const int row  = blockIdx.x * 8 + wave;
    const float* src;
    float* dst;
    if (row < B_N) { src = x + (size_t)row * D_N; dst = xnorm + row; }
    else {
        int r = row - B_N;
        if (r >= K_N) return;
        src = c + (size_t)r * D_N; dst = cnorm + r;
    }
    float4 v = ((const float4*)src)[lane];
    float s = v.x * v.x + v.y * v.y + v.z * v.z + v.w * v.w;
#pragma unroll
    for (int m = 1; m <= 16; m <<= 1) s += __shfl_xor(s, m, 32);
    if (lane == 0) *dst = s;
}

// ------------------------------------------------ dist + softmax (WMMA) --
__global__ void __launch_bounds__(32)
gmm_dist_kernel(const float* __restrict__ x, const float* __restrict__ c,
                const float* __restrict__ xnorm, const float* __restrict__ cnorm,
                float* __restrict__ out, int* __restrict__ assign,
                float* __restrict__ weight) {
    const int lane = threadIdx.x;
    const int g16  = lane >> 4;
    const int l16  = lane & 15;
    const int r0   = blockIdx.x * 16;

    v8f acc[4] = {};
    const float* xrow = x + (size_t)(r0 + l16) * D_N + g16 * 2;
#pragma unroll 4
    for (int s = 0; s < 32; ++s) {
        const int d0 = s * 4;
        v2f a = *(const v2f*)(xrow + d0);
#pragma unroll
        for (int t = 0; t < 4; ++t) {
            v2f b = *(const v2f*)(c + (size_t)(t * 16 + l16) * D_N + d0 + g16 * 2);
            acc[t] = __builtin_amdgcn_wmma_f32_16x16x4_f32(
                false, a, false, b, (short)0, acc[t], false, false);
        }
    }

    float xn[8], cn[4];
#pragma unroll
    for (int i = 0; i < 8; ++i) xn[i] = xnorm[r0 + g16 * 8 + i];
#pragma unroll
    for (int t = 0; t < 4; ++t) cn[t] = cnorm[t * 16 + l16];

    float ex[4][8];
#pragma unroll
    for (int t = 0; t < 4; ++t)
#pragma unroll
        for (int i = 0; i < 8; ++i)
            ex[t][i] = -0.5f * (xn[i] + cn[t] - 2.0f * acc[t][i]);

#pragma unroll
    for (int i = 0; i < 8; ++i) {
        const int row = r0 + g16 * 8 + i;
        float m = fmaxf(fmaxf(ex[0][i], ex[1][i]), fmaxf(ex[2][i], ex[3][i]));
#pragma unroll
        for (int msk = 1; msk <= 8; msk <<= 1) m = fmaxf(m, __shfl_xor(m, msk, 32));
        float s = 0.0f;
#pragma unroll
        for (int t = 0; t < 4; ++t) s += __expf(ex[t][i] - m);
#pragma unroll
        for (int msk = 1; msk <= 8; msk <<= 1) s += __shfl_xor(s, msk, 32);
        const float ls = __logf(s);
#pragma unroll
        for (int t = 0; t < 4; ++t) {
            float lr = ex[t][i] - m - ls;
            lr = fmaxf(lr, -18.420680743952367f);  // log(1e-8)
            out[(size_t)row * K_N + t * 16 + l16] = lr;
        }
        float bv = ex[0][i];
        int bk = l16;
#pragma unroll
        for (int t = 1; t < 4; ++t) {
            if (ex[t][i] > bv) { bv = ex[t][i]; bk = t * 16 + l16; }
        }
#pragma unroll
        for (int msk = 1; msk <= 8; msk <<= 1) {
            float ov = __shfl_xor(bv, msk, 32);
            int   ok = __shfl_xor(bk, msk, 32);
            if (ov > bv || (ov == bv && ok < bk)) { bv = ov; bk = ok; }
        }
        if (l16 == 0) {
            assign[row] = bk;
            atomicAdd(&weight[bk], 1.0f);
        }
    }
}

// ---------------------------------------------------------------- scan ----
__global__ void gmm_scan_kernel(const float* __restrict__ weight,
                                int* __restrict__ offsets) {
    if (threadIdx.x == 0) {
        int acc = 0;
        for (int k = 0; k < K_N; ++k) { offsets[k] = acc; acc += (int)weight[k]; }
        offsets[K_N] = acc;
    }
}

// ------------------------------------------------------------- scatter ---
__global__ void __launch_bounds__(256)
gmm_scatter_kernel(const int* __restrict__ assign, const int* __restrict__ offsets,
                   int* __restrict__ cursor, int* __restrict__ list) {
    int n = blockIdx.x * 256 + threadIdx.x;
    if (n >= B_N) return;
    int k = assign[n];
    int pos = atomicAdd(&cursor[k], 1);
    list[offsets[k] + pos] = n;
}

// ---------------------------------------------------------------- mean ----
__global__ void __launch_bounds__(128)
gmm_mean_kernel(const float* __restrict__ x, const float* __restrict__ c,
                const int* __restrict__ offsets, const int* __restrict__ list,
                const float* __restrict__ weight, float* __restrict__ emp_mean,
                float* __restrict__ mse_out) {
    __shared__ float red[128];
    const int k = blockIdx.x;
    const int d = threadIdx.x;
    const int off = offsets[k];
    const int cnt = offsets[k + 1] - off;
    float s = 0.0f;
    int i = 0;
    for (; i + 3 < cnt; i += 4) {       // 4-way MLP in the gather
        int n0 = list[off + i + 0];
        int n1 = list[off + i + 1];
        int n2 = list[off + i + 2];
        int n3 = list[off + i + 3];
        s += x[(size_t)n0 * D_N + d] + x[(size_t)n1 * D_N + d] +
             x[(size_t)n2 * D_N + d] + x[(size_t)n3 * D_N + d];
    }
    for (; i < cnt; ++i) s += x[(size_t)list[off + i] * D_N + d];

    const float wk = weight[k];
    const float mean = s / (wk + 1e-7f);
    emp_mean[k * D_N + d] = mean;
    const float diff = mean - c[k * D_N + d];
    red[d] = diff * diff;
    __syncthreads();
    for (int stride = 64; stride > 0; stride >>= 1) {
        if (d < stride) red[d] += red[d + stride];
        __syncthreads();
    }
    if (d == 0)
        atomicAdd(mse_out, wk * red[0] / (float)(B_N * D_N));
}

// ------------------- covariance SYRK (WMMA + async double-buffer) --------
__global__ void __launch_bounds__(256)
gmm_covar_kernel(const float* __restrict__ x, const int* __restrict__ offsets,
                 const int* __restrict__ list, const float* __restrict__ weight,
                 const float* __restrict__ emp_mean, float* __restrict__ mse_out) {
    __shared__ __align__(16) float xbuf[2][4 * D_N];
    __shared__ float sdiag[8], soff[8];
    const int k    = blockIdx.x;
    const int tid  = threadIdx.x;
    const int wave = tid >> 5;
    const int lane = tid & 31;
    const int g16  = lane >> 4;
    const int l16  = lane & 15;
    const int off  = offsets[k];
    const int cnt  = offsets[k + 1] - off;
    const float* mean = emp_mean + k * D_N;

    // hoist means used by this lane's fragments
    const int dd = wave * 16 + l16;
    const float md = mean[dd];
    float mb[8];
#pragma unroll
    for (int t = 0; t < 8; ++t) mb[t] = mean[t * 16 + l16];

    // staging slot: each thread async-copies 8 bytes of one row
    const int sr = tid >> 6;          // row 0..3 within chunk
    const int sd = (tid & 63) * 2;    // even column

    const int nfull = cnt >> 2;
    const int rem   = cnt & 3;

    v8f acc[8] = {};

    if (nfull > 0) {   // prologue: chunk 0 in flight
        int n = list[off + sr];
        async_copy8(&x[(size_t)n * D_N + sd], &xbuf[0][sr * D_N + sd]);
    }
    for (int ch = 0; ch < nfull; ++ch) {
        if (ch + 1 < nfull) {          // issue next chunk, then wait for current
            int n = list[off + (ch + 1) * 4 + sr];
            async_copy8(&x[(size_t)n * D_N + sd], &xbuf[(ch + 1) & 1][sr * D_N + sd]);
            wait_async1();
        } else {
            wait_async0();
        }
        __syncthreads();
        const float* buf = xbuf[ch & 1];
        v2f a;
        a.x = buf[(g16 * 2 + 0) * D_N + dd] - md;
        a.y = buf[(g16 * 2 + 1) * D_N + dd] - md;
#pragma unroll
        for (int t = 0; t < 8; ++t) {
            const int ee = t * 16 + l16;
            v2f b;
            b.x = buf[(g16 * 2 + 0) * D_N + ee] - mb[t];
            b.y = buf[(g16 * 2 + 1) * D_N + ee] - mb[t];
            acc[t] = __builtin_amdgcn_wmma_f32_16x16x4_f32(
                false, a, false, b, (short)0, acc[t], false, false);
        }
        __syncthreads();
    }
    if (rem) {   // ragged tail: pad rows with mean[d] so centering yields zero
        float* buf = xbuf[nfull & 1];
#pragma unroll
        for (int j = 0; j < 2; ++j) {
            int flat = tid + j * 256;
            int r = flat >> 7;
            int d = flat & 127;
            float v = mean[d];
            if (r < rem) {
                int n = list[off + nfull * 4 + r];
                v = x[(size_t)n * D_N + d];
            }
            buf[flat] = v;
        }
        __syncthreads();
        v2f a;
        a.x = buf[(g16 * 2 + 0) * D_N + dd] - md;
        a.y = buf[(g16 * 2 + 1) * D_N + dd] - md;
#pragma unroll
        for (int t = 0; t < 8; ++t) {
            const int ee = t * 16 + l16;
            v2f b;
            b.x = buf[(g16 * 2 + 0) * D_N + ee] - mb[t];
            b.y = buf[(g16 * 2 + 1) * D_N + ee] - mb[t];
            acc[t] = __builtin_amdgcn_wmma_f32_16x16x4_f32(
                false, a, false, b, (short)0, acc[t], false, false);
        }
    }

    const float wk  = weight[k];
    const float inv = 1.0f / (wk + 1e-7f);
    float dsum = 0.0f, osum = 0.0f;
#pragma unroll
    for (int t = 0; t < 8; ++t)
#pragma unroll
        for (int i = 0; i < 8; ++i) {
            const int d = wave * 16 + g16 * 8 + i;
            const int e = t * 16 + l16;
            const float cv = acc[t][i] * inv;
            if (d == e) { float z = cv - 1.0f; dsum += z * z; }
            else        { osum += cv * cv; }
        }
#pragma unroll
    for (int m = 1; m <= 16; m <<= 1) {
        dsum += __shfl_xor(dsum, m, 32);
        osum += __shfl_xor(osum, m, 32);
    }
    if (lane == 0) { sdiag[wave] = dsum; soff[wave] = osum; }
    __syncthreads();
    if (tid == 0) {
        float dtot = 0.0f, otot = 0.0f;
        for (int w = 0; w < 8; ++w) { dtot += sdiag[w]; otot += soff[w]; }
        atomicAdd(mse_out,
                  wk * (dtot / (float)(B_N * D_N) +
                        otot / ((float)(B_N * D_N) * 127.0f)));
    }
}

// ------------------------------------------------------------- launch ----
extern "C" void kernel_launch(void* const* d_in, const int* in_sizes, int n_in,
                              void* d_out, int out_size, void* d_ws, size_t ws_size,
                              hipStream_t stream) {
    const float* x = (const float*)d_in[0];   // [B, D]
    const float* c = (const float*)d_in[1];   // [K, D]
    float* out = (float*)d_out;               // [B*K] log_resp + 2 scalars
    char* ws = (char*)d_ws;

    float* xnorm    = (float*)(ws);              // B floats
    float* cnorm    = (float*)(ws + 65536);      // K floats
    int*   assign   = (int*)  (ws + 65792);      // B ints
    float* weight   = (float*)(ws + 131328);     // K floats
    int*   offsets  = (int*)  (ws + 131584);     // K+1 ints
    int*   cursor   = (int*)  (ws + 131844);     // K ints
    int*   list     = (int*)  (ws + 132100);     // B ints
    float* emp_mean = (float*)(ws + 197636);     // K*D floats

    float* scalars = out + (size_t)B_N * K_N;    // [mean_mse, covar_mse]

    hipLaunchKernelGGL(gmm_init_kernel, dim3(1), dim3(64), 0, stream,
                       weight, cursor, scalars);
    hipLaunchKernelGGL(gmm_norms_kernel, dim3((B_N + K_N) / 8), dim3(256), 0, stream,
                       x, c, xnorm, cnorm);
    hipLaunchKernelGGL(gmm_dist_kernel, dim3(B_N / 16), dim3(32), 0, stream,
                       x, c, xnorm, cnorm, out, assign, weight);
    hipLaunchKernelGGL(gmm_scan_kernel, dim3(1), dim3(1), 0, stream,
                       weight, offsets);
    hipLaunchKernelGGL(gmm_scatter_kernel, dim3(B_N / 256), dim3(256), 0, stream,
                       assign, offsets, cursor, list);
    hipLaunchKernelGGL(gmm_mean_kernel, dim3(K_N), dim3(128), 0, stream,
                       x, c, offsets, list, weight, emp_mean, &scalars[0]);
    hipLaunchKernelGGL(gmm_covar_kernel, dim3(K_N), dim3(256), 0, stream,
                       x, offsets, list, weight, emp_mean, &scalars[1]);
}